// TorchJune_75222057222556
// MI455X (gfx1250) — compile-verified
//
#include <hip/hip_runtime.h>
#include <stdint.h>

// ---------------- problem constants (match reference) ----------------
#define G_HH 800000
#define G_CC 20000
#define G_SS 65000
#define G_TOT (G_HH + G_CC + G_SS)

#define BLOCK 256
#define TILE  2048          // agents per block tile (8 per thread)

// per-group record: one 16B line -> one b128 gather per edge type
struct __align__(16) GRec {
  float coef;                 // beta_e * p_contact (static over timesteps)
  unsigned pad;
  unsigned long long acc;     // Q32.32 fixed-point segment sum (per timestep)
};

// ---------------- CDNA5 TDM (tensor_load_to_lds) helpers ----------------
typedef unsigned int u32x4 __attribute__((ext_vector_type(4)));
typedef int          i32x8 __attribute__((ext_vector_type(8)));
typedef int          i32x4 __attribute__((ext_vector_type(4)));

#if defined(__gfx1250__) && __has_builtin(__builtin_amdgcn_tensor_load_to_lds)
#define HAVE_TDM 1
#else
#define HAVE_TDM 0
#endif

__device__ __forceinline__ uint32_t lds_byte_off(const void* p) {
  // generic LDS address keeps the byte offset in the low 32 bits
  return (uint32_t)(uintptr_t)p;
}

// 1-D tile load: DMA `tile_elems` 4-byte elements from gsrc into LDS.
// OOB (x >= oob_elems) reads return zero (ISA 08 §8.2).
__device__ __forceinline__ void tdm_load_1d(uint32_t lds_byte, const void* gsrc,
                                            uint32_t oob_elems, uint32_t tile_elems) {
#if HAVE_TDM
  uint64_t ga = (uint64_t)(uintptr_t)gsrc;
  u32x4 g0;
  g0.x = 1u;                                     // count=1 (valid), is_restore=0, gather=0
  g0.y = lds_byte;                               // lds_addr
  g0.z = (uint32_t)ga;                           // global_addr[31:0]
  g0.w = (uint32_t)((ga >> 32) & 0x01FFFFFFu) | (2u << 30);  // addr[56:32], type=2
  i32x8 g1;
  g1[0] = (int)0x00020000u;                      // data_size=2 (4 bytes)
  g1[1] = (int)((oob_elems & 0xFFFFu) << 16);    // tensor_dim0[15:0]
  g1[2] = (int)((oob_elems >> 16) | (1u << 16)); // tensor_dim0[31:16], tensor_dim1=1
  g1[3] = (int)((tile_elems & 0xFFFFu) << 16);   // tile_dim0
  g1[4] = 1;                                     // tile_dim1=1, tile_dim2=0
  g1[5] = (int)oob_elems;                        // tensor_dim0_stride[31:0]
  g1[6] = 0;
  g1[7] = 0;
  i32x4 z4 = (i32x4){0, 0, 0, 0};                // groups 2/3 unused (<=2D)
  i32x8 z8 = (i32x8){0, 0, 0, 0, 0, 0, 0, 0};    // 6-arg form: extra zero group
  __builtin_amdgcn_tensor_load_to_lds(g0, g1, z4, z4, z8, 0);
#else
  (void)lds_byte; (void)gsrc; (void)oob_elems; (void)tile_elems;
#endif
}

__device__ __forceinline__ void tdm_wait0() {
#if defined(__gfx1250__)
#if __has_builtin(__builtin_amdgcn_s_wait_tensorcnt)
  __builtin_amdgcn_s_wait_tensorcnt(0);
#else
  asm volatile("s_wait_tensorcnt 0x0" ::: "memory");
#endif
#endif
}

// ---------------- setup kernels (run once per launch) ----------------
__global__ void __launch_bounds__(BLOCK)
init_kernel(const float* __restrict__ tr0, const float* __restrict__ su0,
            float* __restrict__ trc, float* __restrict__ suc,
            GRec* __restrict__ rec, int n) {
  int i = blockIdx.x * BLOCK + threadIdx.x;
  if (i < G_TOT) {
    ulonglong2 z; z.x = 0ull; z.y = 0ull;
    *(ulonglong2*)&rec[i] = z;                   // clears coef + acc (count view)
  }
  if (i < n) { trc[i] = tr0[i]; suc[i] = su0[i]; }
}

__global__ void __launch_bounds__(BLOCK)
count_kernel(const int* __restrict__ gh, const int* __restrict__ gc,
             const int* __restrict__ gs, GRec* __restrict__ rec, int n) {
  int i = blockIdx.x * BLOCK + threadIdx.x;
  if (i >= n) return;
  atomicAdd((unsigned*)&rec[gh[i]].acc, 1u);
  atomicAdd((unsigned*)&rec[G_HH + gc[i]].acc, 1u);
  atomicAdd((unsigned*)&rec[G_HH + G_CC + gs[i]].acc, 1u);
}

// coef[g] = beta_e * min(1/(count-1), 1)   (count==1 -> 1/0=+inf -> 1, matches ref)
__global__ void __launch_bounds__(BLOCK)
coef_kernel(GRec* __restrict__ rec, const float* __restrict__ beta) {
  int g = blockIdx.x * BLOCK + threadIdx.x;
  if (g >= G_TOT) return;
  float b = (g < G_HH) ? beta[0] : (g < G_HH + G_CC) ? beta[1] : beta[2];
  float pm1 = (float)((unsigned)rec[g].acc) - 1.0f;
  rec[g].coef = b * fminf(1.0f / pm1, 1.0f);
}

__global__ void __launch_bounds__(BLOCK)
zero_kernel(GRec* __restrict__ rec) {
  int g = blockIdx.x * BLOCK + threadIdx.x;
  if (g < G_TOT) rec[g].acc = 0ull;              // keep coef
}

// ---------------- per-timestep: deterministic fixed-point segment sum ----------------
__global__ void __launch_bounds__(BLOCK)
scatter_kernel(const float* __restrict__ trc, const int* __restrict__ gh,
               const int* __restrict__ gc, const int* __restrict__ gs,
               GRec* __restrict__ rec, int n) {
  __shared__ float sh_t[TILE];
  __shared__ int   sh_h[TILE];
  __shared__ int   sh_c[TILE];
  __shared__ int   sh_s[TILE];
  const int base = blockIdx.x * TILE;
  const uint32_t rem = (uint32_t)(n - base);
#if HAVE_TDM
  if ((threadIdx.x >> 5) == 0) {   // wave 0 issues the DMAs and waits on TENSORcnt
    tdm_load_1d(lds_byte_off(sh_t), trc + base, rem, TILE);
    tdm_load_1d(lds_byte_off(sh_h), gh  + base, rem, TILE);
    tdm_load_1d(lds_byte_off(sh_c), gc  + base, rem, TILE);
    tdm_load_1d(lds_byte_off(sh_s), gs  + base, rem, TILE);
    tdm_wait0();
  }
  asm volatile("" ::: "memory");
  __syncthreads();
#else
  for (int k = threadIdx.x; k < TILE; k += BLOCK) {
    int i = base + k; bool v = i < n;
    sh_t[k] = v ? trc[i] : 0.0f;
    sh_h[k] = v ? gh[i] : 0;
    sh_c[k] = v ? gc[i] : 0;
    sh_s[k] = v ? gs[i] : 0;
  }
  __syncthreads();
#endif
  GRec* rH = rec;
  GRec* rC = rec + G_HH;
  GRec* rS = rec + G_HH + G_CC;
  for (int k = threadIdx.x; k < TILE; k += BLOCK) {
    if (base + k >= n) break;
    // Q32.32 fixed point: order-independent (deterministic) u64 atomics
    unsigned long long q = (unsigned long long)((double)sh_t[k] * 4294967296.0);
    atomicAdd(&rH[sh_h[k]].acc, q);
    atomicAdd(&rC[sh_c[k]].acc, q);
    atomicAdd(&rS[sh_s[k]].acc, q);
  }
}

// ---------------- per-timestep: gather + gumbel hard sample + state update ----------------
__global__ void __launch_bounds__(BLOCK)
gather_kernel(const GRec* __restrict__ rec,
              const int* __restrict__ gh, const int* __restrict__ gc,
              const int* __restrict__ gs,
              const float* __restrict__ gum0, const float* __restrict__ gum1,
              float* __restrict__ trc, float* __restrict__ suc,
              float* __restrict__ out_t, int n) {
  __shared__ int   sh_h[TILE];
  __shared__ int   sh_c[TILE];
  __shared__ int   sh_s[TILE];
  __shared__ float sh_g0[TILE];
  __shared__ float sh_g1[TILE];
  __shared__ float sh_su[TILE];
  __shared__ float sh_tr[TILE];
  const int base = blockIdx.x * TILE;
  const uint32_t rem = (uint32_t)(n - base);
#if HAVE_TDM
  if ((threadIdx.x >> 5) == 0) {
    tdm_load_1d(lds_byte_off(sh_h),  gh   + base, rem, TILE);
    tdm_load_1d(lds_byte_off(sh_c),  gc   + base, rem, TILE);
    tdm_load_1d(lds_byte_off(sh_s),  gs   + base, rem, TILE);
    tdm_load_1d(lds_byte_off(sh_g0), gum0 + base, rem, TILE);
    tdm_load_1d(lds_byte_off(sh_g1), gum1 + base, rem, TILE);
    tdm_load_1d(lds_byte_off(sh_su), suc  + base, rem, TILE);
    tdm_load_1d(lds_byte_off(sh_tr), trc  + base, rem, TILE);
    tdm_wait0();
  }
  asm volatile("" ::: "memory");
  __syncthreads();
#else
  for (int k = threadIdx.x; k < TILE; k += BLOCK) {
    int i = base + k; bool v = i < n;
    sh_h[k]  = v ? gh[i] : 0;
    sh_c[k]  = v ? gc[i] : 0;
    sh_s[k]  = v ? gs[i] : 0;
    sh_g0[k] = v ? gum0[i] : 0.0f;
    sh_g1[k] = v ? gum1[i] : 0.0f;
    sh_su[k] = v ? suc[i] : 0.0f;
    sh_tr[k] = v ? trc[i] : 0.0f;
  }
  __syncthreads();
#endif
  const ulonglong2* rH = (const ulonglong2*)rec;          // one b128 per record
  const ulonglong2* rC = rH + G_HH;
  const ulonglong2* rS = rH + G_HH + G_CC;
  const float SC = 2.3283064365386963e-10f;   // 2^-32
  for (int k = threadIdx.x; k < TILE; k += BLOCK) {
    int i = base + k;
    if (i >= n) break;
    ulonglong2 vh = rH[sh_h[k]];
    ulonglong2 vc = rC[sh_c[k]];
    ulonglong2 vs = rS[sh_s[k]];
    float ts = __uint_as_float((unsigned)vh.x) * ((float)vh.y * SC)
             + __uint_as_float((unsigned)vc.x) * ((float)vc.y * SC)
             + __uint_as_float((unsigned)vs.x) * ((float)vs.y * SC);
    float su = sh_su[k];
    ts *= su;
    float ni = expf(-ts);                         // not_infected
    float l0 = logf(fmaxf(ni, 1e-15f));           // clip(., 1e-15)
    float l1 = logf(fmaxf(1.0f - ni, 1e-15f));
    // argmax over classes; ties -> class 0 (not infected), so strict > for class 1.
    // TAU > 0 does not change the argmax.
    float infected = ((l1 + sh_g1[k]) > (l0 + sh_g0[k])) ? 1.0f : 0.0f;
    out_t[i] = infected;
    trc[i] = sh_tr[k] + 0.2f * infected;
    suc[i] = su - infected;
  }
}

// ---------------- launcher ----------------
extern "C" void kernel_launch(void* const* d_in, const int* in_sizes, int n_in,
                              void* d_out, int out_size, void* d_ws, size_t ws_size,
                              hipStream_t stream) {
  const float* beta   = (const float*)d_in[0];
  const float* tr0    = (const float*)d_in[1];
  const float* su0    = (const float*)d_in[2];
  const float* gumbel = (const float*)d_in[3];
  const int*   gh     = (const int*)d_in[4];
  const int*   gc     = (const int*)d_in[5];
  const int*   gs     = (const int*)d_in[6];
  (void)n_in; (void)ws_size; (void)out_size;

  const int n = in_sizes[1];
  const int T = (int)((long long)in_sizes[3] / (2LL * (long long)n));

  // workspace: rec[G_TOT] (16B each) | transm_cur[n] | susc_cur[n]
  char* ws = (char*)d_ws;
  size_t off = 0;
  GRec* rec = (GRec*)(ws + off);
  off = (off + (size_t)G_TOT * sizeof(GRec) + 255) & ~(size_t)255;
  float* trc = (float*)(ws + off);
  off += (size_t)n * 4;
  float* suc = (float*)(ws + off);

  const int initN   = (n > G_TOT) ? n : G_TOT;
  const int gBlocks = (G_TOT + BLOCK - 1) / BLOCK;
  const int nBlocks = (n + BLOCK - 1) / BLOCK;
  const int tiles   = (n + TILE - 1) / TILE;

  // one-time (per launch): copy state, group counts, fold beta*p_contact into coef
  init_kernel<<<(initN + BLOCK - 1) / BLOCK, BLOCK, 0, stream>>>(tr0, su0, trc, suc, rec, n);
  count_kernel<<<nBlocks, BLOCK, 0, stream>>>(gh, gc, gs, rec, n);
  coef_kernel<<<gBlocks, BLOCK, 0, stream>>>(rec, beta);

  for (int t = 0; t < T; ++t) {
    const float* g0 = gumbel + (size_t)(2 * t) * (size_t)n;
    const float* g1 = g0 + n;
    float* out_t = (float*)d_out + (size_t)t * (size_t)n;
    zero_kernel<<<gBlocks, BLOCK, 0, stream>>>(rec);
    scatter_kernel<<<tiles, BLOCK, 0, stream>>>(trc, gh, gc, gs, rec, n);
    gather_kernel<<<tiles, BLOCK, 0, stream>>>(rec, gh, gc, gs, g0, g1,
                                               trc, suc, out_t, n);
  }
}